// BrainNetwork_13288628814596
// MI455X (gfx1250) — compile-verified
//
#include <hip/hip_runtime.h>
#include <hip/hip_bf16.h>
#include <cmath>

// Problem constants (match reference)
#define RR 64
#define DD 8
#define NN 1024
#define EE 16777216

typedef __attribute__((ext_vector_type(2))) float v2f;
typedef __attribute__((ext_vector_type(8))) float v8f;

// ---------------------------------------------------------------------------
// Phase 1: per-edge gather * weight -> scatter-add into inject[R*N].
// Streams weights/src/dst with 128-bit loads; hist (2MB) gathers hit L2.
// Native f32 global atomics (no CAS loop).
// ---------------------------------------------------------------------------
__global__ __launch_bounds__(256) void edge_scatter_kernel(
    const float* __restrict__ hist,
    const float* __restrict__ weights,
    const int*   __restrict__ src_idx,
    const int*   __restrict__ dst_idx,
    float*       __restrict__ inject)
{
    const int tid = blockIdx.x * blockDim.x + threadIdx.x;   // one int4 per thread
    const int4*   s4 = (const int4*)src_idx;
    const int4*   d4 = (const int4*)dst_idx;
    const float4* w4 = (const float4*)weights;

    int4   s = s4[tid];
    int4   d = d4[tid];
    float4 w = w4[tid];

    unsafeAtomicAdd(&inject[d.x], w.x * hist[s.x]);
    unsafeAtomicAdd(&inject[d.y], w.y * hist[s.y]);
    unsafeAtomicAdd(&inject[d.z], w.z * hist[s.z]);
    unsafeAtomicAdd(&inject[d.w], w.w * hist[s.w]);
}

// ---------------------------------------------------------------------------
// Phase 2: out[r, m] = tanh( sum_n (obs[r,n]+inject[r,n]) * W[r,n,m] )
// One wave32 per (region r, 16-wide m tile). x row staged in LDS.
// GEMV expressed as 256 x V_WMMA_F32_16X16X4_F32:
//   A: row0 = x[n0..n0+3] (rows 1..15 zero) -> only D row 0 is live.
//   B: 4x16 tile of W, one m-column per lane (coalesced dword loads).
// ---------------------------------------------------------------------------
__global__ __launch_bounds__(256) void region_gemv_wmma_kernel(
    const float* __restrict__ obs,
    const float* __restrict__ inject,
    const float* __restrict__ W,
    float*       __restrict__ out)
{
    __shared__ float xs[8][NN];            // 8 waves/block * 1024 f32 = 32 KB

    const int lane  = threadIdx.x & 31;
    const int wv    = threadIdx.x >> 5;
    const int tile  = blockIdx.x * 8 + wv; // 4096 tiles = 64 regions * 64 m-tiles
    const int r     = tile >> 6;
    const int m0    = (tile & 63) << 4;

    // Stage x = obs + inject for this wave's region.
    {
        const float* o = obs    + r * NN;
        const float* j = inject + r * NN;
        for (int n = lane; n < NN; n += 32)
            xs[wv][n] = o[n] + j[n];
    }
    __syncthreads();

    const int   grp   = lane >> 4;                 // lane half (K pairing)
    const int   laneM = lane & 15;                 // output column within tile
    const float amask = (laneM == 0) ? 1.0f : 0.0f;

    // Lane's fixed column in W[r]: rows advance by N floats.
    // A layout (ISA 16x4 f32): lanes 0-15 -> K0,K1 ; lanes 16-31 -> K2,K3.
    // B layout (4x16 f32):     lanes 0-15 -> rows K0,K1 ; lanes 16-31 -> K2,K3.
    const float* Wr = W + ((size_t)((r * NN) + 2 * grp)) * NN + (m0 + laneM);

    v8f acc = {0.f, 0.f, 0.f, 0.f, 0.f, 0.f, 0.f, 0.f};

    for (int n0 = 0; n0 < NN; n0 += 4) {
        v2f a, b;
        a.x = amask * xs[wv][n0 + 2 * grp];
        a.y = amask * xs[wv][n0 + 2 * grp + 1];
        b.x = Wr[(size_t)n0 * NN];            // W[r][n0 + 2*grp    ][m]
        b.y = Wr[(size_t)n0 * NN + NN];       // W[r][n0 + 2*grp + 1][m]
        // D = A*B + C ; only row 0 of D is live (A rows 1..15 are zero).
        acc = __builtin_amdgcn_wmma_f32_16x16x4_f32(
            /*neg_a=*/false, a, /*neg_b=*/false, b,
            /*c_mod=*/(short)0, acc, /*reuse_a=*/false, /*reuse_b=*/false);
    }

    // D row M=0 lives in acc[0] on lanes 0..15 (N = lane).
    if (lane < 16) {
        out[r * NN + m0 + laneM] = tanhf(acc[0]);
    }
}

// ---------------------------------------------------------------------------
// Launch: memset(inject) -> edge scatter -> fused add+GEMV+tanh
// ---------------------------------------------------------------------------
extern "C" void kernel_launch(void* const* d_in, const int* in_sizes, int n_in,
                              void* d_out, int out_size, void* d_ws, size_t ws_size,
                              hipStream_t stream) {
    const float* hist    = (const float*)d_in[0];  // [R, D, N]
    const float* obs     = (const float*)d_in[1];  // [R, N]
    const float* weights = (const float*)d_in[2];  // [E]
    const float* W       = (const float*)d_in[3];  // [R, N, N]
    const int*   src_idx = (const int*)d_in[4];    // [E]
    const int*   dst_idx = (const int*)d_in[5];    // [E]
    float*       out     = (float*)d_out;          // [R*N]
    float*       inject  = (float*)d_ws;           // [R*N] scratch accumulator

    (void)in_sizes; (void)n_in; (void)out_size; (void)ws_size;

    // Zero the scatter accumulator (capture-safe async memset node).
    hipMemsetAsync(inject, 0, (size_t)RR * NN * sizeof(float), stream);

    // Edge phase: E/4 threads, 4 edges each via 128-bit loads.
    const int edge_threads = EE / 4;                 // 4,194,304
    edge_scatter_kernel<<<edge_threads / 256, 256, 0, stream>>>(
        hist, weights, src_idx, dst_idx, inject);

    // Region GEMV phase: 4096 wave-tiles, 8 waves per 256-thread block.
    region_gemv_wmma_kernel<<<(RR * 64) / 8, 256, 0, stream>>>(
        obs, inject, W, out);
}